// CustomGCN_45990509805904
// MI455X (gfx1250) — compile-verified
//
#include <hip/hip_runtime.h>
#include <math.h>

typedef __attribute__((ext_vector_type(2))) float v2f;
typedef __attribute__((ext_vector_type(4))) float v4f;
typedef __attribute__((ext_vector_type(8))) float v8f;

#define N_NODES 100000
#define N_EDGES 1600000
#define FEAT    64
#define EMBED   128
#define NCLS    40

// ---------------------------------------------------------------- zero fill
// n must be a multiple of 4 (true for N*64 and N*40); b128 stores.
__global__ __launch_bounds__(256) void gcn_zero4(float* __restrict__ p, long n4) {
  long i = (long)blockIdx.x * blockDim.x + threadIdx.x;
  if (i < n4) ((v4f*)p)[i] = (v4f){0.0f, 0.0f, 0.0f, 0.0f};
}

// ------------------------------------------------- SPMM: edge-parallel scatter
// agg[dst, f..f+3] += w[e] * feat[src, f..f+3]
// One thread per (edge, 4 features): one b128 gather + 4 f32 atomics.
// NF must be a multiple of 4 so the v4f gathers stay 16B-aligned.
template <int NF>
__global__ __launch_bounds__(256) void gcn_spmm_scatter4(
    const float* __restrict__ feat, const int* __restrict__ src,
    const int* __restrict__ dst, const float* __restrict__ w,
    float* __restrict__ out, long total) {
  constexpr int QP = NF / 4;                 // quads per edge
  long gid = (long)blockIdx.x * blockDim.x + threadIdx.x;
  if (gid >= total) return;
  int e = (int)(gid / QP);
  int f = (int)(gid % QP) * 4;
  int s = src[e];
  int d = dst[e];
  float ww = w[e];
  v4f v = *(const v4f*)(feat + (long)s * NF + f);   // global_load_b128
  float* o = out + (long)d * NF + f;
  atomicAdd(o + 0, ww * v.x);
  atomicAdd(o + 1, ww * v.y);
  atomicAdd(o + 2, ww * v.z);
  atomicAdd(o + 3, ww * v.w);
}

// -------------------------------------------------- dense GEMM via f32 WMMA
// C[M x N] = A[M x K] * B[K x N], one wave32 per 16x16 output tile.
// A layout (16x4, 32-bit): lane m = lane&15, klo = 2*(lane>>4), holds K=klo,klo+1
// B layout (4x16, 32-bit): lane n = lane&15, same klo pairing
// C/D layout: VGPR r -> M = r + 8*(lane>>4), N = lane&15
template <int K, int N, int NT, bool RELU, bool NMASK>
__global__ __launch_bounds__(256) void gcn_wmma_gemm(
    const float* __restrict__ A, const float* __restrict__ B,
    float* __restrict__ C, int totalTiles) {
  const int wave = threadIdx.x >> 5;
  const int lane = threadIdx.x & 31;
  const int tile = blockIdx.x * 8 + wave;
  if (tile >= totalTiles) return;          // wave-uniform exit; EXEC stays all-1s

  const int mt = tile / NT;
  const int nt = tile % NT;
  const int m    = lane & 15;
  const int half = lane >> 4;              // 0 or 1
  const int klo  = half * 2;

  const int row  = mt * 16 + m;            // M is an exact multiple of 16
  const int ncol = nt * 16 + m;            // B column for this lane (n = lane&15)

  // Mask out-of-range B columns with a select + multiply (no EXEC divergence).
  const int   bcol  = (!NMASK || ncol < N) ? ncol : (N - 1);
  const float bmask = (!NMASK || ncol < N) ? 1.0f : 0.0f;

  const float* __restrict__ arow = A + (long)row * K;

  v8f c = {};
#pragma unroll
  for (int k = 0; k < K; k += 4) {
    v2f a = *(const v2f*)(arow + k + klo);             // 8B-aligned pair load
    v2f b;
    b.x = B[(long)(k + klo)     * N + bcol] * bmask;
    b.y = B[(long)(k + klo + 1) * N + bcol] * bmask;
    // 8 args: (neg_a, A, neg_b, B, c_mod, C, reuse_a, reuse_b)
    c = __builtin_amdgcn_wmma_f32_16x16x4_f32(
        false, a, false, b, (short)0, c, false, false);
  }

#pragma unroll
  for (int r = 0; r < 8; ++r) {
    int rr = mt * 16 + r + half * 8;
    int cc = nt * 16 + (lane & 15);
    if (!NMASK || cc < N) {
      float v = c[r];
      if (RELU) v = fmaxf(v, 0.0f);
      C[(long)rr * N + cc] = v;
    }
  }
}

// ------------------------------------------------------- row-wise log_softmax
__global__ __launch_bounds__(256) void gcn_log_softmax(
    const float* __restrict__ in, float* __restrict__ out, int n) {
  int i = blockIdx.x * blockDim.x + threadIdx.x;
  if (i >= n) return;
  const float* __restrict__ row = in + (long)i * NCLS;
  float mx = -INFINITY;
#pragma unroll
  for (int c = 0; c < NCLS; ++c) mx = fmaxf(mx, row[c]);
  float s = 0.0f;
#pragma unroll
  for (int c = 0; c < NCLS; ++c) s += __expf(row[c] - mx);
  float lse = mx + __logf(s);
  float* __restrict__ orow = out + (long)i * NCLS;
#pragma unroll
  for (int c = 0; c < NCLS; ++c) orow[c] = row[c] - lse;
}

// --------------------------------------------------------------------- launch
extern "C" void kernel_launch(void* const* d_in, const int* in_sizes, int n_in,
                              void* d_out, int out_size, void* d_ws, size_t ws_size,
                              hipStream_t stream) {
  const float* x    = (const float*)d_in[0];   // [N, 64]
  const float* W1   = (const float*)d_in[1];   // [64, 128]
  const float* W2   = (const float*)d_in[2];   // [128, 40]
  const float* ew   = (const float*)d_in[3];   // [E]
  const int*   esrc = (const int*)d_in[4];     // [E]
  const int*   edst = (const int*)d_in[5];     // [E]
  float*       out  = (float*)d_out;           // [N, 40]

  // Workspace layout (all f32): agg1 | h | t | agg2
  float* agg1 = (float*)d_ws;                                  // N*64
  float* h    = agg1 + (size_t)N_NODES * FEAT;                 // N*128
  float* t    = h    + (size_t)N_NODES * EMBED;                // N*40
  float* agg2 = t    + (size_t)N_NODES * NCLS;                 // N*40

  const int TPB = 256;

  // 1) zero the atomic accumulators (graph replays don't re-poison)
  {
    long n1 = (long)N_NODES * FEAT / 4;
    gcn_zero4<<<(int)((n1 + TPB - 1) / TPB), TPB, 0, stream>>>(agg1, n1);
    long n2 = (long)N_NODES * NCLS / 4;
    gcn_zero4<<<(int)((n2 + TPB - 1) / TPB), TPB, 0, stream>>>(agg2, n2);
  }

  // 2) agg1 = A @ x   (scatter-add over edges, 4 feats/thread)
  {
    long total = (long)N_EDGES * (FEAT / 4);
    gcn_spmm_scatter4<FEAT><<<(int)((total + TPB - 1) / TPB), TPB, 0, stream>>>(
        x, esrc, edst, ew, agg1, total);
  }

  // 3) h = relu(agg1 @ W1)   [100000x64]@[64x128], WMMA f32 16x16x4
  {
    const int NT = EMBED / 16;                   // 8
    int tiles = (N_NODES / 16) * NT;             // 6250*8 = 50000
    gcn_wmma_gemm<FEAT, EMBED, NT, true, false>
        <<<(tiles + 7) / 8, TPB, 0, stream>>>(agg1, W1, h, tiles);
  }

  // 4) t = h @ W2   [100000x128]@[128x40], N padded to 3 tiles, masked
  {
    const int NT = (NCLS + 15) / 16;             // 3
    int tiles = (N_NODES / 16) * NT;             // 18750
    gcn_wmma_gemm<EMBED, NCLS, NT, false, true>
        <<<(tiles + 7) / 8, TPB, 0, stream>>>(h, W2, t, tiles);
  }

  // 5) agg2 = A @ t   (scatter-add over edges, 4 feats/thread)
  {
    long total = (long)N_EDGES * (NCLS / 4);
    gcn_spmm_scatter4<NCLS><<<(int)((total + TPB - 1) / TPB), TPB, 0, stream>>>(
        t, esrc, edst, ew, agg2, total);
  }

  // 6) out = log_softmax(agg2, axis=1)
  gcn_log_softmax<<<(N_NODES + TPB - 1) / TPB, TPB, 0, stream>>>(agg2, out, N_NODES);
}